// BaseQuantizer_55671366091043
// MI455X (gfx1250) — compile-verified
//
#include <hip/hip_runtime.h>

#define ALPHA 0.95f

typedef float v2f __attribute__((ext_vector_type(2)));
typedef float v8f __attribute__((ext_vector_type(8)));

constexpr int D = 512;      // feature dim
constexpr int N = 8192;     // codebook size
constexpr int MTILE = 16;   // tokens per block

// ---------------------------------------------------------------- init
__global__ __launch_bounds__(256) void vq_init(const float* __restrict__ freq,
                                               float* __restrict__ out_freq,
                                               float* __restrict__ ws_loss) {
    int i = blockIdx.x * 256 + threadIdx.x;
    if (i < N) out_freq[i] = ALPHA * freq[i];
    if (i == 0) *ws_loss = 0.0f;
}

// ---------------------------------------------------------------- ||cb_n||^2
__global__ __launch_bounds__(256) void vq_cbnorm(const float* __restrict__ cb,
                                                 float* __restrict__ cbnorm) {
    int n = blockIdx.x * 256 + threadIdx.x;
    float s = 0.0f;
#pragma unroll 8
    for (int d = 0; d < D; ++d) {
        float c = cb[(size_t)d * N + n];
        s += c * c;
    }
    cbnorm[n] = s;
}

// ---------------------------------------------------------------- argmin via FP32 WMMA
// Block: 256 threads (8 waves). Each block owns 16 tokens; each wave owns a
// 1024-code slab of the codebook and accumulates 16x16 C tiles with
// v_wmma_f32_16x16x4_f32 over K=512.
__global__ __launch_bounds__(256) void vq_argmin(const float* __restrict__ x,
                                                 const float* __restrict__ cb,
                                                 const float* __restrict__ cbnorm,
                                                 int* __restrict__ ws_idx,
                                                 float* __restrict__ out_idx) {
    __shared__ float lds_a[MTILE * D];   // 32 KB: A tile, row-major [row][k]
    __shared__ float redMin[8][16];
    __shared__ int   redIdx[8][16];

    const int tid  = threadIdx.x;
    const int wave = tid >> 5;
    const int lane = tid & 31;
    const int t0   = blockIdx.x * MTILE;

    // cooperative load of 16x512 A tile (float4, fully coalesced)
    for (int i = tid; i < (MTILE * D) / 4; i += 256) {
        int e   = i * 4;
        int row = e / D;
        int k   = e % D;
        *(float4*)(&lds_a[e]) = *(const float4*)(&x[(size_t)(t0 + row) * D + k]);
    }
    __syncthreads();

    // ISA 16x16x4 f32 fragment layout:
    //  A: lane<16 -> row=lane, K={k0,k0+1}; lane>=16 -> row=lane-16, K={k0+2,k0+3}
    //  B: lane<16 -> col=lane, K={k0,k0+1}; lane>=16 -> col=lane-16, K={k0+2,k0+3}
    //  C vgpr r: lane<16 -> M=r, N=lane; lane>=16 -> M=r+8, N=lane-16
    const int half    = lane >> 4;        // 0 | 1
    const int l16     = lane & 15;
    const int rowBase = half << 3;        // 0 | 8
    const float* aRow = &lds_a[l16 * D + (half << 1)];

    float runMin[8];
    int   runIdx[8];
#pragma unroll
    for (int r = 0; r < 8; ++r) { runMin[r] = 3.4e38f; runIdx[r] = 0; }

    const int nStart = wave * (N / 8);
    for (int n0 = nStart; n0 < nStart + N / 8; n0 += 16) {
        v8f acc = {0.f, 0.f, 0.f, 0.f, 0.f, 0.f, 0.f, 0.f};
        const float* bCol = &cb[(size_t)(half << 1) * N + (n0 + l16)];
#pragma unroll 4
        for (int k0 = 0; k0 < D; k0 += 4) {
            v2f a = *(const v2f*)(aRow + k0);                 // ds_load_b64
            v2f b;
            b.x = bCol[(size_t)k0 * N];                       // lane-coalesced, L2-resident
            b.y = bCol[(size_t)(k0 + 1) * N];
            acc = __builtin_amdgcn_wmma_f32_16x16x4_f32(
                false, a, false, b, (short)0, acc, false, false);
        }
        // dist = ||cb||^2 - 2*dot  (||x||^2 constant per row -> dropped)
        float cbn = cbnorm[n0 + l16];
        int   n   = n0 + l16;
#pragma unroll
        for (int r = 0; r < 8; ++r) {
            float dv = cbn - 2.0f * acc[r];
            if (dv < runMin[r]) { runMin[r] = dv; runIdx[r] = n; }
        }
    }

    // reduce across the 16 lanes of each half (masks < 16 keep halves separate)
#pragma unroll
    for (int r = 0; r < 8; ++r) {
        float v = runMin[r];
        int   i = runIdx[r];
#pragma unroll
        for (int m = 1; m < 16; m <<= 1) {
            float ov = __shfl_xor(v, m, 32);
            int   oi = __shfl_xor(i, m, 32);
            if (ov < v || (ov == v && oi < i)) { v = ov; i = oi; }
        }
        if (l16 == 0) { redMin[wave][rowBase + r] = v; redIdx[wave][rowBase + r] = i; }
    }
    __syncthreads();

    // final reduce across waves; first-index tie-break (wave slabs are ordered in n)
    if (tid < 16) {
        float v = redMin[0][tid];
        int   i = redIdx[0][tid];
#pragma unroll
        for (int w = 1; w < 8; ++w) {
            float ov = redMin[w][tid];
            int   oi = redIdx[w][tid];
            if (ov < v || (ov == v && oi < i)) { v = ov; i = oi; }
        }
        int token = t0 + tid;
        ws_idx[token]  = i;
        out_idx[token] = (float)i;
    }
}

// ---------------------------------------------------------------- dequant + loss + counts
__global__ __launch_bounds__(256) void vq_dequant(const float* __restrict__ x,
                                                  const float* __restrict__ cb,
                                                  const int* __restrict__ ws_idx,
                                                  float* __restrict__ out_x,
                                                  float* __restrict__ out_freq,
                                                  float* __restrict__ ws_loss) {
    __shared__ float wsum[8];
    const int token = blockIdx.x;
    const int tid   = threadIdx.x;
    const int idx   = ws_idx[token];

    float s = 0.0f;
#pragma unroll
    for (int d = tid; d < D; d += 256) {
        float q  = cb[(size_t)d * N + idx];      // gather column (L2-resident)
        float xv = x[(size_t)token * D + d];
        out_x[(size_t)token * D + d] = q;        // x_st == x_quantized exactly
        float diff = q - xv;
        s += diff * diff;
    }
#pragma unroll
    for (int m = 16; m >= 1; m >>= 1) s += __shfl_xor(s, m, 32);
    if ((tid & 31) == 0) wsum[tid >> 5] = s;
    __syncthreads();
    if (tid == 0) {
        float t = 0.0f;
#pragma unroll
        for (int w = 0; w < 8; ++w) t += wsum[w];
        atomicAdd(ws_loss, t);
        atomicAdd(&out_freq[idx], 1.0f - ALPHA); // EMA one-hot count
    }
}

// ---------------------------------------------------------------- finalize loss
__global__ void vq_finalize(const float* __restrict__ ws_loss,
                            float* __restrict__ out_loss, float invCount) {
    // inner_loss = commitment + codebook loss = 2 * mean((xq - x)^2)
    *out_loss = 2.0f * (*ws_loss) * invCount;
}

// ---------------------------------------------------------------- launch
extern "C" void kernel_launch(void* const* d_in, const int* in_sizes, int n_in,
                              void* d_out, int out_size, void* d_ws, size_t ws_size,
                              hipStream_t stream) {
    const float* x    = (const float*)d_in[0];   // [B,L,D]
    const float* cb   = (const float*)d_in[1];   // [D,N]
    const float* freq = (const float*)d_in[2];   // [N]

    const int tokens = in_sizes[0] / D;          // B*L = 16384

    float* out      = (float*)d_out;
    float* out_x    = out;                                   // [tokens*D]
    float* out_idx  = out + (size_t)tokens * D;              // [tokens]
    float* out_loss = out_idx + tokens;                      // [1]
    float* out_freq = out_loss + 1;                          // [N]

    float* ws_cbnorm = (float*)d_ws;                         // N floats
    int*   ws_idx    = (int*)(ws_cbnorm + N);                // tokens ints
    float* ws_loss   = (float*)(ws_idx + tokens);            // 1 float

    vq_init    <<<(N + 255) / 256, 256, 0, stream>>>(freq, out_freq, ws_loss);
    vq_cbnorm  <<<N / 256,        256, 0, stream>>>(cb, ws_cbnorm);
    vq_argmin  <<<tokens / MTILE, 256, 0, stream>>>(x, cb, ws_cbnorm, ws_idx, out_idx);
    vq_dequant <<<tokens,         256, 0, stream>>>(x, cb, ws_idx, out_x, out_freq, ws_loss);
    vq_finalize<<<1, 1, 0, stream>>>(ws_loss, out_loss, 1.0f / ((float)tokens * (float)D));
}